// VectorQuantizer_12618613915787
// MI455X (gfx1250) — compile-verified
//
#include <hip/hip_runtime.h>
#include <hip/hip_bf16.h>

// ---------------------------------------------------------------------------
// VQ-VAE vector quantizer for MI455X (gfx1250, wave32).
//   z:        [131072, 64] fp32      codebook: [1024, 64] fp32
//   out:  quantized [131072*64] fp32 | indices-as-float [131072] | loss [1]
//
// Distance GEMM on V_WMMA_F32_16X16X4_F32 (exact fp32 -> safe argmin).
// Codebook + ||e||^2 staged in 260KB LDS (CDNA5: 320KB/WGP).
// TWO independent WMMA accumulator chains per loop iteration (2 x 16-code
// tiles) so the matrix pipe is not serialized on C-accumulation latency.
// ---------------------------------------------------------------------------

typedef __attribute__((ext_vector_type(2))) float v2f;
typedef __attribute__((ext_vector_type(4))) float v4f;
typedef __attribute__((ext_vector_type(8))) float v8f;

#define NUM_CODES 1024
#define DIM 64
#define N_TOTAL (32 * 4096)          // 131072 rows
#define ROWS_PER_WAVE 16
#define WAVES_PER_BLOCK 8
#define THREADS 256
#define LDS_CB_FLOATS (NUM_CODES * DIM)              // 65536 floats = 256KB
#define LDS_TOTAL_FLOATS (LDS_CB_FLOATS + NUM_CODES) // + 1024 norms

extern "C" __global__ void vq_main_kernel(const float* __restrict__ z,
                                          const float* __restrict__ codebook,
                                          float* __restrict__ outQ,
                                          float* __restrict__ outIdx,
                                          float* __restrict__ partial)
{
    extern __shared__ float lds[];
    float* ldsCB   = lds;                 // [1024][64]
    float* ldsNorm = lds + LDS_CB_FLOATS; // [1024]

    const int tid = threadIdx.x;

    // --- Stage entire codebook into LDS (float4, coalesced) -----------------
    {
        const v4f* src = (const v4f*)codebook;
        v4f*       dst = (v4f*)ldsCB;
        #pragma unroll 4
        for (int i = tid; i < LDS_CB_FLOATS / 4; i += THREADS)
            dst[i] = src[i];
    }
    __syncthreads();

    // --- ||e||^2 per code (from LDS) ---------------------------------------
    for (int k = tid; k < NUM_CODES; k += THREADS) {
        const float* row = ldsCB + k * DIM;
        float s = 0.f;
        #pragma unroll
        for (int j = 0; j < DIM; ++j) s = fmaf(row[j], row[j], s);
        ldsNorm[k] = s;
    }
    __syncthreads();

    const int lane = tid & 31;
    const int wave = tid >> 5;
    const int rowTile = blockIdx.x * WAVES_PER_BLOCK + wave;
    const int row0 = rowTile * ROWS_PER_WAVE;

    // A-operand layout for V_WMMA_F32_16X16X4_F32 (16x4, MxK):
    //   lanes 0-15 hold M=lane, VGPR0/1 = K=0/1; lanes 16-31 = K=2/3.
    const int m  = lane & 15;
    const int kh = (lane >> 4) * 2;

    v2f a[16];   // 16 k-steps cover D=64; lives in registers for the scan
    {
        const float* zr = z + (size_t)(row0 + m) * DIM + kh;
        #pragma unroll
        for (int s = 0; s < 16; ++s)
            a[s] = *(const v2f*)(zr + 4 * s);
    }

    float bestVal[8];
    int   bestIdx[8];
    #pragma unroll
    for (int r = 0; r < 8; ++r) { bestVal[r] = 3.4e38f; bestIdx[r] = 0; }

    const int n = m;   // B column (code within tile) owned by this lane

    // --- Scan 1024 codes: 32 per iteration, 2 independent WMMA chains ------
    for (int c = 0; c < NUM_CODES; c += 32) {
        v8f acc0 = {};
        v8f acc1 = {};
        const float* cbr0 = ldsCB + (size_t)(c + n) * DIM + kh;        // tile 0
        const float* cbr1 = ldsCB + (size_t)(c + 16 + n) * DIM + kh;   // tile 1
        #pragma unroll
        for (int s = 0; s < 16; ++s) {
            v2f b0 = *(const v2f*)(cbr0 + 4 * s);       // ds_load_b64
            v2f b1 = *(const v2f*)(cbr1 + 4 * s);
            acc0 = __builtin_amdgcn_wmma_f32_16x16x4_f32(
                       false, a[s], false, b0, (short)0, acc0, false, false);
            acc1 = __builtin_amdgcn_wmma_f32_16x16x4_f32(
                       false, a[s], false, b1, (short)0, acc1, false, false);
        }
        // score = ||e||^2 - 2*(z.e)   (||z||^2 constant per row -> dropped)
        const float en0 = ldsNorm[c + n];
        const float en1 = ldsNorm[c + 16 + n];
        const int   i0  = c + n;
        const int   i1  = c + 16 + n;
        #pragma unroll
        for (int r = 0; r < 8; ++r) {
            float v0 = fmaf(-2.f, acc0[r], en0);
            if (v0 < bestVal[r]) { bestVal[r] = v0; bestIdx[r] = i0; }
            float v1 = fmaf(-2.f, acc1[r], en1);
            if (v1 < bestVal[r]) { bestVal[r] = v1; bestIdx[r] = i1; }
        }
    }

    // --- Argmin across the 16 lanes of each half (rows r / r+8) ------------
    #pragma unroll
    for (int r = 0; r < 8; ++r) {
        float v = bestVal[r];
        int   i = bestIdx[r];
        #pragma unroll
        for (int msk = 1; msk < 16; msk <<= 1) {
            float ov = __shfl_xor(v, msk, 32);
            int   oi = __shfl_xor(i, msk, 32);
            if (ov < v || (ov == v && oi < i)) { v = ov; i = oi; }
        }
        bestVal[r] = v;
        bestIdx[r] = i;
    }

    // Lanes 0 / 16 own rows row0+0..7 / row0+8..15: write indices (as float)
    if (m == 0) {
        const int base = row0 + (lane >> 4) * 8;
        #pragma unroll
        for (int r = 0; r < 8; ++r)
            outIdx[base + r] = (float)bestIdx[r];
    }

    // --- Gather quantized rows from LDS + accumulate loss partial ----------
    float lsum = 0.f;
    #pragma unroll
    for (int mm = 0; mm < 16; ++mm) {
        const int srcLane = (mm >> 3) * 16;                 // lane 0 or 16
        const int k   = __shfl(bestIdx[mm & 7], srcLane, 32);
        const int row = row0 + mm;
        v2f q = *(const v2f*)(ldsCB + (size_t)k * DIM + lane * 2);
        const float* zp = z + (size_t)row * DIM + lane * 2;
        const float d0 = q.x - zp[0];
        const float d1 = q.y - zp[1];
        lsum = fmaf(d0, d0, lsum);
        lsum = fmaf(d1, d1, lsum);
        *(v2f*)(outQ + (size_t)row * DIM + lane * 2) = q;   // global_store_b64
    }
    #pragma unroll
    for (int msk = 1; msk < 32; msk <<= 1)
        lsum += __shfl_xor(lsum, msk, 32);
    if (lane == 0) partial[rowTile] = lsum;
}

// Deterministic final reduction: fixed strided sums + fixed LDS tree.
extern "C" __global__ void vq_loss_reduce_kernel(const float* __restrict__ partial,
                                                 float* __restrict__ outLoss,
                                                 int n, float scale)
{
    __shared__ float red[256];
    float s = 0.f;
    for (int i = threadIdx.x; i < n; i += 256) s += partial[i];
    red[threadIdx.x] = s;
    __syncthreads();
    for (int off = 128; off > 0; off >>= 1) {
        if (threadIdx.x < off) red[threadIdx.x] += red[threadIdx.x + off];
        __syncthreads();
    }
    if (threadIdx.x == 0) *outLoss = red[0] * scale;
}

extern "C" void kernel_launch(void* const* d_in, const int* in_sizes, int n_in,
                              void* d_out, int out_size, void* d_ws, size_t ws_size,
                              hipStream_t stream)
{
    (void)in_sizes; (void)n_in; (void)out_size; (void)ws_size;

    const float* z        = (const float*)d_in[0];
    const float* codebook = (const float*)d_in[1];

    float* outQ    = (float*)d_out;                       // 131072*64
    float* outIdx  = outQ + (size_t)N_TOTAL * DIM;        // 131072
    float* outLoss = outIdx + N_TOTAL;                    // 1
    float* partial = (float*)d_ws;                        // 8192 floats

    const int numWaves = N_TOTAL / ROWS_PER_WAVE;         // 8192
    const dim3 grid(numWaves / WAVES_PER_BLOCK);          // 1024 blocks
    const size_t shmem = LDS_TOTAL_FLOATS * sizeof(float); // 266,240 B (<320KB WGP)

    vq_main_kernel<<<grid, THREADS, shmem, stream>>>(z, codebook, outQ, outIdx, partial);
    vq_loss_reduce_kernel<<<1, 256, 0, stream>>>(
        partial, outLoss, numWaves, 1.25f / ((float)N_TOTAL * (float)DIM));
}